// Con_Proximity_14877766713743
// MI455X (gfx1250) — compile-verified
//
#include <hip/hip_runtime.h>
#include <stdint.h>

typedef __attribute__((ext_vector_type(2))) float v2f;
typedef __attribute__((ext_vector_type(8))) float v8f;

#define B_ROWS 32768
#define C_CLS  43
#define D_FEAT 1024
#define NPAD   48          // 43 -> 3 tiles of 16
#define KCHUNK 64          // K staged through LDS per iteration
#define MTILE  128         // rows per workgroup (8 waves x 16)
#define XPITCH 68          // floats; multiple of 4 (16B align for b128), breaks bank conflicts
#define CPITCH 68
#define NSTAGES (D_FEAT / KCHUNK)   // 16
#define ASYNC_OPS_PER_STAGE 11      // 8 x-tiles + 3 center-tiles per wave

// Async DMA: global -> LDS, 16B per lane, tracked by ASYNCcnt (CDNA5).
__device__ __forceinline__ void async_copy_b128(uint32_t lds_off, uint64_t gaddr) {
    asm volatile("global_load_async_to_lds_b128 %0, %1, off"
                 :: "v"(lds_off), "v"(gaddr) : "memory");
}

__global__ __launch_bounds__(256) void conprox_main_kernel(
    const float* __restrict__ x, const float* __restrict__ centers,
    const int* __restrict__ labels, float* __restrict__ partials)
{
    __shared__ __align__(16) float xs[2][MTILE * XPITCH];  // double-buffered x tile
    __shared__ __align__(16) float cs[2][NPAD * CPITCH];   // double-buffered centers tile
    __shared__ float wsum[8];

    const int t    = threadIdx.x;
    const int wave = t >> 5;
    const int lane = t & 31;
    const int lo   = lane & 15;
    const int hi   = lane >> 4;
    const int blockRow = blockIdx.x * MTILE;

    // Issue one stage's async copies (EXEC-uniform: 11 ops per wave, always).
    auto issue_stage = [&](int s) {
        const int kbase = s * KCHUNK;
        float* xsb = xs[s & 1];
        float* csb = cs[s & 1];
        #pragma unroll
        for (int i = 0; i < 8; ++i) {          // 128 rows x 64 k = 2048 float4, 8/thread
            int c = t + i * 256;
            int row = c >> 4, kq = c & 15;
            async_copy_b128((uint32_t)(uintptr_t)(xsb + row * XPITCH + kq * 4),
                            (uint64_t)(uintptr_t)(x + (size_t)(blockRow + row) * D_FEAT
                                                    + kbase + kq * 4));
        }
        #pragma unroll
        for (int i = 0; i < 3; ++i) {          // 48 rows x 64 k = 768 float4, 3/thread
            int c = t + i * 256;
            int j = c >> 4, kq = c & 15;
            int jj = (j < C_CLS) ? j : 0;      // clamp source: no divergence; cols >=43 masked later
            async_copy_b128((uint32_t)(uintptr_t)(csb + j * CPITCH + kq * 4),
                            (uint64_t)(uintptr_t)(centers + (size_t)jj * D_FEAT
                                                    + kbase + kq * 4));
        }
    };

    v8f acc0 = {}, acc1 = {}, acc2 = {};
    float xsqp = 0.0f, csq0p = 0.0f, csq1p = 0.0f, csq2p = 0.0f;

    issue_stage(0);   // prologue

    for (int s = 0; s < NSTAGES; ++s) {
        if (s + 1 < NSTAGES) {
            issue_stage(s + 1);   // prefetch next stage into other buffer
            // 11 new ops outstanding on top of stage s's 11; async loads retire in
            // order, so waiting <=11 guarantees stage s's data has landed in LDS.
            asm volatile("s_wait_asynccnt 11" ::: "memory");
        } else {
            asm volatile("s_wait_asynccnt 0" ::: "memory");
        }
        __syncthreads();   // all waves' stage-s tiles visible

        // ---- compute: each wave -> 16 rows x 48 cols, K in steps of 4 ----
        // A 16x4 f32 layout: lane l holds x[m = l&15][k0 + 2*(l>>4) + {0,1}]
        // B 4x16 layout (C/D lo/hi convention): lane l holds c[j = l&15][k0 + 2*(l>>4) + {0,1}]
        const float* xsb  = xs[s & 1];
        const float* csb  = cs[s & 1];
        const float* xrow = xsb + (wave * 16 + lo) * XPITCH;
        #pragma unroll
        for (int kk = 0; kk < KCHUNK; kk += 4) {
            v2f a  = *(const v2f*)(xrow + kk + 2 * hi);
            v2f b0 = *(const v2f*)(csb + ( 0 + lo) * CPITCH + kk + 2 * hi);
            v2f b1 = *(const v2f*)(csb + (16 + lo) * CPITCH + kk + 2 * hi);
            v2f b2 = *(const v2f*)(csb + (32 + lo) * CPITCH + kk + 2 * hi);
            // norms for free from the fragment stream (VALU co-executes with XDL WMMA)
            xsqp  += a.x * a.x + a.y * a.y;
            csq0p += b0.x * b0.x + b0.y * b0.y;
            csq1p += b1.x * b1.x + b1.y * b1.y;
            csq2p += b2.x * b2.x + b2.y * b2.y;
            acc0 = __builtin_amdgcn_wmma_f32_16x16x4_f32(false, a, false, b0, (short)0, acc0, false, false);
            acc1 = __builtin_amdgcn_wmma_f32_16x16x4_f32(false, a, false, b1, (short)0, acc1, false, false);
            acc2 = __builtin_amdgcn_wmma_f32_16x16x4_f32(false, a, false, b2, (short)0, acc2, false, false);
        }
        __syncthreads();   // buf[s&1] free before it is re-targeted at iter s+1's issue
    }

    // Combine the two half-K slices: lane l then holds ||x[wave*16 + (l&15)]||^2 etc.
    float xsqv = xsqp  + __shfl_xor(xsqp,  16, 32);
    float csq0 = csq0p + __shfl_xor(csq0p, 16, 32);
    float csq1 = csq1p + __shfl_xor(csq1p, 16, 32);
    float csq2 = csq2p + __shfl_xor(csq2p, 16, 32);

    // ---- epilogue: d = ||x||^2 + ||c||^2 - 2*dot, clamp, mask, sum ----
    // C/D layout: VGPR r: lanes 0-15 -> M=r, N=lane; lanes 16-31 -> M=r+8, N=lane-16
    float part = 0.0f;
    #pragma unroll
    for (int r = 0; r < 8; ++r) {
        int mloc = wave * 16 + r + 8 * hi;
        int mg   = blockRow + mloc;
        int lab  = labels[mg];
        float xq = __shfl(xsqv, r + 8 * hi, 32);   // norm of this accumulator row
        float d;
        d = xq + csq0 - 2.0f * acc0[r];
        d = fminf(fmaxf(d, 1e-12f), 1e12f);
        if (lo != lab) part += d;                          // j = lo (< 43)
        d = xq + csq1 - 2.0f * acc1[r];
        d = fminf(fmaxf(d, 1e-12f), 1e12f);
        if (16 + lo != lab) part += d;                     // j = 16+lo (< 43)
        d = xq + csq2 - 2.0f * acc2[r];
        d = fminf(fmaxf(d, 1e-12f), 1e12f);
        if (32 + lo < C_CLS && 32 + lo != lab) part += d;  // j = 32+lo, mask padding
    }

    // wave32 reduction, then cross-wave via LDS
    for (int off = 16; off > 0; off >>= 1)
        part += __shfl_down(part, off, 32);
    if (lane == 0) wsum[wave] = part;
    __syncthreads();
    if (t == 0) {
        float sacc = 0.0f;
        #pragma unroll
        for (int w = 0; w < 8; ++w) sacc += wsum[w];
        partials[blockIdx.x] = sacc;
    }
}

__global__ __launch_bounds__(256) void conprox_reduce_kernel(
    const float* __restrict__ partials, float* __restrict__ out)
{
    __shared__ float sh[256];
    sh[threadIdx.x] = partials[threadIdx.x];
    __syncthreads();
    for (int s = 128; s > 0; s >>= 1) {
        if (threadIdx.x < s) sh[threadIdx.x] += sh[threadIdx.x + s];
        __syncthreads();
    }
    if (threadIdx.x == 0)
        out[0] = sh[0] * (1.0f / (32768.0f * 42.0f));   // 1 / (B*(C-1))
}

extern "C" void kernel_launch(void* const* d_in, const int* in_sizes, int n_in,
                              void* d_out, int out_size, void* d_ws, size_t ws_size,
                              hipStream_t stream)
{
    const float* x       = (const float*)d_in[0];
    const float* centers = (const float*)d_in[1];
    const int*   labels  = (const int*)d_in[2];
    float* out      = (float*)d_out;
    float* partials = (float*)d_ws;   // 256 floats of scratch

    conprox_main_kernel<<<B_ROWS / MTILE, 256, 0, stream>>>(x, centers, labels, partials);
    conprox_reduce_kernel<<<1, 256, 0, stream>>>(partials, out);
}